// GAT_DGL_44994077393442
// MI455X (gfx1250) — compile-verified
//
#include <hip/hip_runtime.h>
#include <math.h>

typedef float v2f __attribute__((ext_vector_type(2)));
typedef float v8f __attribute__((ext_vector_type(8)));

#define HID 32
#define NEG_SLOPE 0.2f

// ---------------------------------------------------------------------------
// GEMM: H[N,32] = X[N,K] @ W[K,32] via V_WMMA_F32_16X16X4_F32 (fp32-exact).
// One wave32 per 16-row tile; two 16x16 f32 accumulators (cols 0-15, 16-31).
// A layout: lane&15 = M row, (lane>>4)*2 = K pair   -> v2f per step
// B layout: lane&15 = N col, (lane>>4)*2 = K rows   -> 2 strided W loads
// C layout: vgpr r -> M = r (+8 for lanes 16-31), N = lane&15
// ---------------------------------------------------------------------------
template <int K>
__global__ __launch_bounds__(256)
void gat_gemm_wmma(const float* __restrict__ X, const float* __restrict__ W,
                   float* __restrict__ H, int N) {
  const int wave  = (int)((blockIdx.x * blockDim.x + threadIdx.x) >> 5);
  const int lane  = (int)(threadIdx.x & 31);
  const int tiles = (N + 15) >> 4;
  if (wave >= tiles) return;                 // wave-uniform: EXEC stays all-1s
  const int row0 = wave << 4;
  const int m    = lane & 15;
  const int kg   = (lane >> 4) << 1;         // 0 or 2
  int rowA = row0 + m; if (rowA > N - 1) rowA = N - 1;   // tail clamp (loads)
  const float* xrow = X + (size_t)rowA * K + kg;
  const int col = lane & 15;

  v8f c0 = {0.f,0.f,0.f,0.f,0.f,0.f,0.f,0.f};
  v8f c1 = {0.f,0.f,0.f,0.f,0.f,0.f,0.f,0.f};

#pragma unroll 4
  for (int k0 = 0; k0 < K; k0 += 4) {
    v2f a = *(const v2f*)(xrow + k0);                    // 8B aligned
    const float* wb = W + (size_t)(k0 + kg) * HID;
    v2f b0, b1;
    b0.x = wb[col];       b0.y = wb[HID + col];
    b1.x = wb[16 + col];  b1.y = wb[HID + 16 + col];
    c0 = __builtin_amdgcn_wmma_f32_16x16x4_f32(false, a, false, b0,
                                               (short)0, c0, false, false);
    c1 = __builtin_amdgcn_wmma_f32_16x16x4_f32(false, a, false, b1,
                                               (short)0, c1, false, false);
  }

  const int rbase = row0 + ((lane >> 4) << 3);
  float* out = H + (size_t)rbase * HID + col;
  if (row0 + 16 <= N) {
    // full tile (always taken for N % 16 == 0): branch-free clauseable stores
#pragma unroll
    for (int r = 0; r < 8; ++r) {
      out[(size_t)r * HID]      = c0[r];
      out[(size_t)r * HID + 16] = c1[r];
    }
  } else {
#pragma unroll
    for (int r = 0; r < 8; ++r) {
      if (rbase + r < N) {
        out[(size_t)r * HID]      = c0[r];
        out[(size_t)r * HID + 16] = c1[r];
      }
    }
  }
}

// ---------------------------------------------------------------------------
// Per-node attention logits: el[n] = h[n,:].al, er[n] = h[n,:].ar
// One wave32 per node, shuffle-tree reduction.
// ---------------------------------------------------------------------------
__global__ __launch_bounds__(256)
void node_dots(const float* __restrict__ h, const float* __restrict__ al,
               const float* __restrict__ ar, float* __restrict__ el,
               float* __restrict__ er, int N) {
  const int node = (int)((blockIdx.x * blockDim.x + threadIdx.x) >> 5);
  const int lane = (int)(threadIdx.x & 31);
  if (node >= N) return;
  float hv = h[(size_t)node * HID + lane];
  float a = hv * al[lane];
  float b = hv * ar[lane];
#pragma unroll
  for (int off = 16; off > 0; off >>= 1) {
    a += __shfl_xor(a, off, 32);
    b += __shfl_xor(b, off, 32);
  }
  if (lane == 0) { el[node] = a; er[node] = b; }
}

// ---------------------------------------------------------------------------
__global__ __launch_bounds__(256)
void seg_init(float* __restrict__ emax, float* __restrict__ denom,
              float* __restrict__ agg, int N) {
  const size_t i = (size_t)blockIdx.x * blockDim.x + threadIdx.x;
  if (i < (size_t)N) { emax[i] = -__builtin_inff(); denom[i] = 0.f; }
  if (i < (size_t)N * HID) agg[i] = 0.f;
}

// Sign-split float atomic max: positives via signed-int max, negatives via
// unsigned-int min (monotone bit orderings). Init must be -inf.
__device__ __forceinline__ void atomicMaxF32(float* addr, float v) {
  if (v >= 0.f) atomicMax((int*)addr, __float_as_int(v));
  else          atomicMin((unsigned int*)addr, __float_as_uint(v));
}

__global__ __launch_bounds__(256)
void edge_score(const int* __restrict__ src, const int* __restrict__ dst,
                const float* __restrict__ el, const float* __restrict__ er,
                float* __restrict__ sbuf, float* __restrict__ emax, int E) {
  const int e = (int)(blockIdx.x * blockDim.x + threadIdx.x);
  if (e >= E) return;
  float v = el[src[e]] + er[dst[e]];
  v = (v > 0.f) ? v : NEG_SLOPE * v;          // LeakyReLU
  sbuf[e] = v;
  atomicMaxF32(&emax[dst[e]], v);
}

__global__ __launch_bounds__(256)
void edge_exp(const int* __restrict__ dst, float* __restrict__ sbuf,
              const float* __restrict__ emax, float* __restrict__ denom,
              int E) {
  const int e = (int)(blockIdx.x * blockDim.x + threadIdx.x);
  if (e >= E) return;
  const int d = dst[e];
  float v = expf(sbuf[e] - emax[d]);
  sbuf[e] = v;
  atomicAdd(&denom[d], v);
}

// 32 lanes per edge: lanes sweep the 32 feature channels (coalesced gather of
// h[src,:] and coalesced atomic scatter-add into agg[dst,:], both L2-resident).
__global__ __launch_bounds__(256)
void edge_agg(const int* __restrict__ src, const int* __restrict__ dst,
              const float* __restrict__ sbuf, const float* __restrict__ denom,
              const float* __restrict__ h, float* __restrict__ agg, int E) {
  const size_t t = (size_t)blockIdx.x * blockDim.x + threadIdx.x;
  const int e    = (int)(t >> 5);
  const int lane = (int)(t & 31);
  if (e >= E) return;
  const int s = src[e], d = dst[e];
  const float alpha = sbuf[e] / denom[d];
  atomicAdd(&agg[(size_t)d * HID + lane], alpha * h[(size_t)s * HID + lane]);
}

__global__ __launch_bounds__(256)
void finalize_elu(const float* __restrict__ agg, const float* __restrict__ bias,
                  float* __restrict__ out, size_t total) {
  const size_t i = (size_t)blockIdx.x * blockDim.x + threadIdx.x;
  if (i >= total) return;
  float v = agg[i] + bias[i & 31];
  out[i] = (v > 0.f) ? v : (expf(v) - 1.0f);  // ELU alpha=1
}

__global__ __launch_bounds__(256)
void finalize_bias(const float* __restrict__ agg, const float* __restrict__ bias,
                   float* __restrict__ out, size_t total) {
  const size_t i = (size_t)blockIdx.x * blockDim.x + threadIdx.x;
  if (i >= total) return;
  out[i] = agg[i] + bias[i & 31];
}

// ---------------------------------------------------------------------------
extern "C" void kernel_launch(void* const* d_in, const int* in_sizes, int n_in,
                              void* d_out, int out_size, void* d_ws, size_t ws_size,
                              hipStream_t stream) {
  const float* features = (const float*)d_in[0];
  const int*   src      = (const int*)  d_in[1];
  const int*   dst      = (const int*)  d_in[2];
  const float* W1  = (const float*)d_in[3];
  const float* al1 = (const float*)d_in[4];
  const float* ar1 = (const float*)d_in[5];
  const float* b1  = (const float*)d_in[6];
  const float* W2  = (const float*)d_in[7];
  const float* al2 = (const float*)d_in[8];
  const float* ar2 = (const float*)d_in[9];
  const float* b2  = (const float*)d_in[10];
  const int N = in_sizes[0] / 128;   // features is [N,128]
  const int E = in_sizes[1];

  // workspace layout (all f32): ~ (100*N + E)*4 bytes ≈ 46.4 MB
  float* ws    = (float*)d_ws;
  float* h     = ws;  ws += (size_t)N * HID;
  float* agg   = ws;  ws += (size_t)N * HID;
  float* hact  = ws;  ws += (size_t)N * HID;
  float* el    = ws;  ws += N;
  float* er    = ws;  ws += N;
  float* emax  = ws;  ws += N;
  float* denom = ws;  ws += N;
  float* sbuf  = ws;  ws += E;

  const int BLK = 256;
  const int tiles = (N + 15) / 16;
  const unsigned gGemm  = (unsigned)(((size_t)tiles * 32 + BLK - 1) / BLK);
  const unsigned gNW    = (unsigned)(((size_t)N * HID + BLK - 1) / BLK);  // N*32 threads
  const unsigned gEdge  = (unsigned)(((size_t)E + BLK - 1) / BLK);
  const unsigned gEdgeW = (unsigned)(((size_t)E * 32 + BLK - 1) / BLK);

  // ---- layer 1: GATConv(features; W1) + ELU ----
  gat_gemm_wmma<128><<<gGemm, BLK, 0, stream>>>(features, W1, h, N);
  node_dots<<<gNW, BLK, 0, stream>>>(h, al1, ar1, el, er, N);
  seg_init<<<gNW, BLK, 0, stream>>>(emax, denom, agg, N);
  edge_score<<<gEdge, BLK, 0, stream>>>(src, dst, el, er, sbuf, emax, E);
  edge_exp<<<gEdge, BLK, 0, stream>>>(dst, sbuf, emax, denom, E);
  edge_agg<<<gEdgeW, BLK, 0, stream>>>(src, dst, sbuf, denom, h, agg, E);
  finalize_elu<<<gNW, BLK, 0, stream>>>(agg, b1, hact, (size_t)N * HID);

  // ---- layer 2: GATConv(hact; W2) ----
  gat_gemm_wmma<32><<<gGemm, BLK, 0, stream>>>(hact, W2, h, N);
  node_dots<<<gNW, BLK, 0, stream>>>(h, al2, ar2, el, er, N);
  seg_init<<<gNW, BLK, 0, stream>>>(emax, denom, agg, N);
  edge_score<<<gEdge, BLK, 0, stream>>>(src, dst, el, er, sbuf, emax, E);
  edge_exp<<<gEdge, BLK, 0, stream>>>(dst, sbuf, emax, denom, E);
  edge_agg<<<gEdgeW, BLK, 0, stream>>>(src, dst, sbuf, denom, h, agg, E);
  finalize_bias<<<gNW, BLK, 0, stream>>>(agg, b2, (float*)d_out, (size_t)N * HID);
}